// GradientFusionloss_16965120819861
// MI455X (gfx1250) — compile-verified
//
#include <hip/hip_runtime.h>

// ---------------------------------------------------------------------------
// Sobel-gradient fusion loss, MI455X (gfx1250, wave32).
//   loss = mean(| grad(gen_y) - max(grad(vis_y), grad(ir)) |)
// Memory-bound: ~100MB read @ 23.3 TB/s -> ~4.3us floor. One streaming pass,
// float4 row loads + register reuse, WMMA-based wave reduction.
// ---------------------------------------------------------------------------

typedef __attribute__((ext_vector_type(2))) float v2f;
typedef __attribute__((ext_vector_type(8))) float v8f;

#define BATCH   32
#define HH      512
#define WW      512
#define HWSZ    (HH * WW)                 // 262144
#define NPIX    (BATCH * HWSZ)            // 8388608
#define GROUPS  (NPIX / 4)                // 2097152 four-pixel groups
#define BLOCK   256
#define NBLOCKS (GROUPS / BLOCK)          // 8192, exact

// Sum s over all 32 lanes of the wave using V_WMMA_F32_16X16X4_F32.
// A layout (16x4 f32): lanes 0-15 supply K=0 (vgpr0) / K=1 (vgpr1);
// lanes 16-31 supply K=2 / K=3. With a = {s, 0} and B = ones:
//   D[m][n] = s[m] + s[m+16]   (replicated over n)
// D layout: vgpr j, lanes 0-15 hold M=j, lanes 16-31 hold M=j+8, so
// sum(d[0..7]) gives half-sums; one xor-16 shuffle completes the total.
__device__ __forceinline__ float wave_sum32(float s) {
    v2f a; a[0] = s;    a[1] = 0.0f;
    v2f b; b[0] = 1.0f; b[1] = 1.0f;
    v8f c = {};
    v8f d = __builtin_amdgcn_wmma_f32_16x16x4_f32(
        /*neg_a=*/false, a, /*neg_b=*/false, b,
        /*c_mod=*/(short)0, c, /*reuse_a=*/false, /*reuse_b=*/false);
    float t = d[0] + d[1] + d[2] + d[3] + d[4] + d[5] + d[6] + d[7];
    t += __shfl_xor(t, 16, 32);
    return t;  // total in every lane
}

// Load 6 contiguous row values [w0-1, w0+4] with zero padding outside [0,WW).
__device__ __forceinline__ void load_row6(const float* __restrict__ row,
                                          int w0, bool valid, float r[6]) {
    if (valid) {
        const float4 v = *reinterpret_cast<const float4*>(row + w0);  // aligned
        r[0] = (w0 > 0)       ? row[w0 - 1] : 0.0f;
        r[1] = v.x; r[2] = v.y; r[3] = v.z; r[4] = v.w;
        r[5] = (w0 + 4 < WW)  ? row[w0 + 4] : 0.0f;
    } else {
        r[0] = r[1] = r[2] = r[3] = r[4] = r[5] = 0.0f;
    }
}

// Sobel magnitude |gx|+|gy| (cross-correlation, zero pad) for 4 pixels.
__device__ __forceinline__ void grad4(const float* __restrict__ plane,
                                      int h, int w0, float g[4]) {
    float t[6], m[6], bo[6];
    const int hu = (h > 0)      ? h - 1 : 0;
    const int hd = (h < HH - 1) ? h + 1 : HH - 1;
    load_row6(plane + hu * WW, w0, h > 0,      t);
    load_row6(plane + h  * WW, w0, true,       m);
    load_row6(plane + hd * WW, w0, h < HH - 1, bo);
#pragma unroll
    for (int i = 0; i < 4; ++i) {
        // SOBEL_X = [[-1,0,1],[-2,0,2],[-1,0,1]]
        float gx = (t[i + 2] - t[i]) + 2.0f * (m[i + 2] - m[i]) + (bo[i + 2] - bo[i]);
        // SOBEL_Y = [[1,2,1],[0,0,0],[-1,-2,-1]]
        float gy = (t[i] + 2.0f * t[i + 1] + t[i + 2])
                 - (bo[i] + 2.0f * bo[i + 1] + bo[i + 2]);
        g[i] = fabsf(gx) + fabsf(gy);
    }
}

__global__ void __launch_bounds__(BLOCK)
sobel_loss_partial(const float* __restrict__ vis,   // (32,3,512,512)
                   const float* __restrict__ ir,    // (32,1,512,512)
                   const float* __restrict__ gen,   // (32,3,512,512)
                   float* __restrict__ partial) {   // NBLOCKS floats
    const int g   = blockIdx.x * BLOCK + threadIdx.x;  // four-pixel group id
    const int w0  = (g & (WW / 4 - 1)) << 2;           // 0..508 step 4
    const int row = g >> 7;                            // g / 128
    const int h   = row & (HH - 1);
    const int b   = row >> 9;                          // row / 512

    const float* visP = vis + (size_t)b * 3 * HWSZ;    // channel 0
    const float* genP = gen + (size_t)b * 3 * HWSZ;    // channel 0
    const float* irP  = ir  + (size_t)b * HWSZ;

    float gv[4], gi[4], gg[4];
    grad4(visP, h, w0, gv);
    grad4(irP,  h, w0, gi);
    grad4(genP, h, w0, gg);

    float s = 0.0f;
#pragma unroll
    for (int i = 0; i < 4; ++i)
        s += fabsf(gg[i] - fmaxf(gv[i], gi[i]));

    const float waveTot = wave_sum32(s);               // v_wmma_f32_16x16x4_f32

    __shared__ float wsum[BLOCK / 32];
    const int lane = threadIdx.x & 31;
    const int wave = threadIdx.x >> 5;
    if (lane == 0) wsum[wave] = waveTot;
    __syncthreads();
    if (threadIdx.x == 0) {
        float tot = 0.0f;
#pragma unroll
        for (int i = 0; i < BLOCK / 32; ++i) tot += wsum[i];
        partial[blockIdx.x] = tot;                     // overwritten every call
    }
}

__global__ void __launch_bounds__(BLOCK)
sobel_loss_final(const float* __restrict__ partial, float* __restrict__ out) {
    float s = 0.0f;
    for (int i = threadIdx.x; i < NBLOCKS; i += BLOCK) s += partial[i];

    const float waveTot = wave_sum32(s);               // v_wmma_f32_16x16x4_f32

    __shared__ float wsum[BLOCK / 32];
    const int lane = threadIdx.x & 31;
    const int wave = threadIdx.x >> 5;
    if (lane == 0) wsum[wave] = waveTot;
    __syncthreads();
    if (threadIdx.x == 0) {
        float tot = 0.0f;
#pragma unroll
        for (int i = 0; i < BLOCK / 32; ++i) tot += wsum[i];
        out[0] = tot * (1.0f / (float)NPIX);           // mean
    }
}

extern "C" void kernel_launch(void* const* d_in, const int* in_sizes, int n_in,
                              void* d_out, int out_size, void* d_ws, size_t ws_size,
                              hipStream_t stream) {
    (void)in_sizes; (void)n_in; (void)out_size; (void)ws_size;
    const float* vis = (const float*)d_in[0];   // (32,3,512,512) f32
    const float* ir  = (const float*)d_in[1];   // (32,1,512,512) f32
    const float* gen = (const float*)d_in[2];   // (32,3,512,512) f32
    float* out       = (float*)d_out;           // scalar f32
    float* partial   = (float*)d_ws;            // NBLOCKS * 4 bytes scratch

    sobel_loss_partial<<<NBLOCKS, BLOCK, 0, stream>>>(vis, ir, gen, partial);
    sobel_loss_final<<<1, BLOCK, 0, stream>>>(partial, out);
}